// EdgeFeatureAttention_60919816126982
// MI455X (gfx1250) — compile-verified
//
#include <hip/hip_runtime.h>
#include <hip/hip_bf16.h>
#include <math.h>

#define N_NODES 100000
#define N_EDGES 1600000
#define IN_DIM  128
#define EDGE_DIM 16
#define NH 8           // heads
#define HD 16          // dim per head
#define SLOPE 0.2f

typedef __attribute__((ext_vector_type(2))) float v2f;
typedef __attribute__((ext_vector_type(8))) float v8f;

// ---------------------------------------------------------------------------
// K1: fold a_edge into W_edge:  V[h][k] = sum_d W_edge[(h*16+d)*16 + k] * a_edge[h][d]
// attn_vec is (8, 48): [:, :16]=a_src, [:,16:32]=a_dst, [:,32:48]=a_edge
// ---------------------------------------------------------------------------
__global__ void fold_edge_attn(const float* __restrict__ W_edge,
                               const float* __restrict__ attn_vec,
                               float* __restrict__ V) {
    int t = threadIdx.x;            // 0..127
    int h = t >> 4, k = t & 15;
    float acc = 0.f;
#pragma unroll
    for (int d = 0; d < HD; ++d)
        acc += W_edge[(h * HD + d) * EDGE_DIM + k] * attn_vec[h * 48 + 32 + d];
    V[t] = acc;
}

// ---------------------------------------------------------------------------
// K2: node GEMM h = X @ W^T via V_WMMA_F32_16X16X4_F32, fused with the
// per-head attention dots s_src / s_dst (16-lane shuffle reductions).
// grid.x = N/16 row-tiles, block = 256 (8 waves), wave w == head w.
//
// A (16x4 f32): lane l holds row M=l%16, K = {2*(l/16), 2*(l/16)+1}  (2 VGPRs)
// B (4x16 f32): lane l holds col N=l%16, K = {2*(l/16), 2*(l/16)+1}; since
//   B'[k,n] = W[n,k], the B fragment is a contiguous float2 from W's row n.
// C (16x16 f32, 8 VGPRs): VGPR r -> M = r + 8*(l/16), N = l%16.
// ---------------------------------------------------------------------------
__global__ __launch_bounds__(256) void node_gemm_wmma(
    const float* __restrict__ X,        // N x 128
    const float* __restrict__ W,        // 128 x 128
    const float* __restrict__ attn_vec, // 8 x 48
    float* __restrict__ h_ws,           // N x 128
    float* __restrict__ s_src,          // N x 8
    float* __restrict__ s_dst) {        // N x 8
    const int hh   = threadIdx.x >> 5;  // head / wave id
    const int lane = threadIdx.x & 31;
    const int half = lane >> 4;
    const int mrow = lane & 15;
    const int m0   = blockIdx.x * 16;
    const int n0   = hh * 16;

    const float* arow = X + (size_t)(m0 + mrow) * IN_DIM + 2 * half;
    const float* brow = W + (size_t)(n0 + mrow) * IN_DIM + 2 * half;

    v8f c = {};
#pragma unroll
    for (int k = 0; k < IN_DIM; k += 4) {
        v2f a = *reinterpret_cast<const v2f*>(arow + k);
        v2f b = *reinterpret_cast<const v2f*>(brow + k);
        c = __builtin_amdgcn_wmma_f32_16x16x4_f32(
                /*neg_a=*/false, a, /*neg_b=*/false, b,
                /*c_mod=*/(short)0, c, /*reuse_a=*/false, /*reuse_b=*/false);
    }

    const float a_s = attn_vec[hh * 48 + mrow];        // a_src[hh, d=mrow]
    const float a_d = attn_vec[hh * 48 + 16 + mrow];   // a_dst[hh, d=mrow]

#pragma unroll
    for (int r = 0; r < 8; ++r) {
        const int m = m0 + r + 8 * half;
        h_ws[(size_t)m * IN_DIM + n0 + mrow] = c[r];
        float ps = c[r] * a_s;
        float pd = c[r] * a_d;
#pragma unroll
        for (int off = 8; off >= 1; off >>= 1) {       // reduce within 16-lane half
            ps += __shfl_xor(ps, off, 32);
            pd += __shfl_xor(pd, off, 32);
        }
        if (mrow == 0) {
            s_src[m * NH + hh] = ps;
            s_dst[m * NH + hh] = pd;
        }
    }
}

// ---------------------------------------------------------------------------
// K3: per-edge score + leaky relu + segment max (int-punned atomicMax,
// exact for seg_max initialized to 0.0f == bit pattern 0).
// ---------------------------------------------------------------------------
__global__ __launch_bounds__(256) void edge_score(
    const float* __restrict__ EF,       // E x 16
    const int*   __restrict__ eidx,     // [2, E] flat
    const float* __restrict__ V,        // 8 x 16 (folded)
    const float* __restrict__ s_src,    // N x 8
    const float* __restrict__ s_dst,    // N x 8
    float* __restrict__ score_ws,       // E x 8
    float* __restrict__ seg_max) {      // N x 8 (init 0)
    __shared__ float Vs[128];
    if (threadIdx.x < 128) Vs[threadIdx.x] = V[threadIdx.x];
    __syncthreads();

    const int e = blockIdx.x * blockDim.x + threadIdx.x;
    if (e >= N_EDGES) return;
    const int src = eidx[e];
    const int dst = eidx[N_EDGES + e];

    float ef[16];
    const float4* efp = reinterpret_cast<const float4*>(EF + (size_t)e * EDGE_DIM);
#pragma unroll
    for (int i = 0; i < 4; ++i) {
        float4 v = efp[i];
        ef[4*i] = v.x; ef[4*i+1] = v.y; ef[4*i+2] = v.z; ef[4*i+3] = v.w;
    }

    float sc8[8];
#pragma unroll
    for (int h = 0; h < NH; ++h) {
        float se = 0.f;
#pragma unroll
        for (int k = 0; k < 16; ++k) se = fmaf(ef[k], Vs[h * 16 + k], se);
        float sc = s_src[src * NH + h] + s_dst[dst * NH + h] + se;
        sc = (sc >= 0.f) ? sc : SLOPE * sc;            // leaky relu
        sc8[h] = sc;
        atomicMax(reinterpret_cast<int*>(seg_max) + dst * NH + h,
                  __float_as_int(sc));
    }
    float4* sp = reinterpret_cast<float4*>(score_ws + (size_t)e * NH);
    sp[0] = make_float4(sc8[0], sc8[1], sc8[2], sc8[3]);
    sp[1] = make_float4(sc8[4], sc8[5], sc8[6], sc8[7]);
}

// ---------------------------------------------------------------------------
// K4: exp(score - seg_max[dst]) and segment-sum denominator.
// ---------------------------------------------------------------------------
__global__ __launch_bounds__(256) void edge_exp(
    const int*   __restrict__ eidx,
    const float* __restrict__ seg_max,  // N x 8
    float* __restrict__ score_ws,       // E x 8 (in: score, out: exp)
    float* __restrict__ denom) {        // N x 8 (init 0)
    const int e = blockIdx.x * blockDim.x + threadIdx.x;
    if (e >= N_EDGES) return;
    const int dst = eidx[N_EDGES + e];

    float4* sp = reinterpret_cast<float4*>(score_ws + (size_t)e * NH);
    float4 s0 = sp[0], s1 = sp[1];
    float sc[8] = {s0.x, s0.y, s0.z, s0.w, s1.x, s1.y, s1.z, s1.w};
#pragma unroll
    for (int h = 0; h < NH; ++h) {
        float ex = __expf(sc[h] - seg_max[dst * NH + h]);
        sc[h] = ex;
        atomicAdd(denom + dst * NH + h, ex);
    }
    sp[0] = make_float4(sc[0], sc[1], sc[2], sc[3]);
    sp[1] = make_float4(sc[4], sc[5], sc[6], sc[7]);
}

// ---------------------------------------------------------------------------
// K5: message pass. Thread per (edge, channel). Gather h[src], scale by
// normalized attention weight, atomic-add into out[dst].
// ---------------------------------------------------------------------------
__global__ __launch_bounds__(256) void edge_message(
    const int*   __restrict__ eidx,
    const float* __restrict__ h_ws,     // N x 128
    const float* __restrict__ exp_ws,   // E x 8
    const float* __restrict__ denom,    // N x 8
    float* __restrict__ out) {          // N x 128 (init 0)
    const int t = blockIdx.x * blockDim.x + threadIdx.x;   // < E*128 = 204.8M
    const int e  = t >> 7;
    const int cc = t & 127;
    const int hh = cc >> 4;
    const int src = eidx[e];
    const int dst = eidx[N_EDGES + e];
    const float w = exp_ws[(size_t)e * NH + hh] /
                    (denom[dst * NH + hh] + 1e-9f);
    atomicAdd(out + (size_t)dst * IN_DIM + cc,
              h_ws[(size_t)src * IN_DIM + cc] * w);
}

// ---------------------------------------------------------------------------
// K6: ELU in place on d_out.
// ---------------------------------------------------------------------------
__global__ void elu_kernel(float* __restrict__ out, int n) {
    const int i = blockIdx.x * blockDim.x + threadIdx.x;
    if (i < n) {
        float x = out[i];
        out[i] = (x > 0.f) ? x : (__expf(x) - 1.f);
    }
}

// ---------------------------------------------------------------------------
extern "C" void kernel_launch(void* const* d_in, const int* in_sizes, int n_in,
                              void* d_out, int out_size, void* d_ws, size_t ws_size,
                              hipStream_t stream) {
    const float* node_emb  = (const float*)d_in[0];   // N x 128
    const float* edge_feat = (const float*)d_in[1];   // E x 16
    const float* W_node    = (const float*)d_in[2];   // 128 x 128
    const float* W_edge    = (const float*)d_in[3];   // 128 x 16
    const float* attn_vec  = (const float*)d_in[4];   // 8 x 48
    const int*   edge_idx  = (const int*)d_in[5];     // 2 x E
    float* out = (float*)d_out;                       // N x 128

    // workspace layout (floats)
    float* ws = (float*)d_ws;
    float* h_ws     = ws;                       ws += (size_t)N_NODES * IN_DIM; // 12.8M
    float* s_src    = ws;                       ws += (size_t)N_NODES * NH;     // 0.8M
    float* s_dst    = ws;                       ws += (size_t)N_NODES * NH;
    float* seg_max  = ws;                       ws += (size_t)N_NODES * NH;
    float* denom    = ws;                       ws += (size_t)N_NODES * NH;
    float* Vfold    = ws;                       ws += 128;
    float* score_ws = ws;                       ws += (size_t)N_EDGES * NH;     // 12.8M

    // init accumulators (0.0f bit pattern == 0)
    hipMemsetAsync(out,     0, (size_t)N_NODES * IN_DIM * sizeof(float), stream);
    hipMemsetAsync(seg_max, 0, (size_t)N_NODES * NH * sizeof(float), stream);
    hipMemsetAsync(denom,   0, (size_t)N_NODES * NH * sizeof(float), stream);

    fold_edge_attn<<<1, 128, 0, stream>>>(W_edge, attn_vec, Vfold);

    node_gemm_wmma<<<N_NODES / 16, 256, 0, stream>>>(
        node_emb, W_node, attn_vec, h_ws, s_src, s_dst);

    const int eb = (N_EDGES + 255) / 256;
    edge_score<<<eb, 256, 0, stream>>>(edge_feat, edge_idx, Vfold,
                                       s_src, s_dst, score_ws, seg_max);
    edge_exp<<<eb, 256, 0, stream>>>(edge_idx, seg_max, score_ws, denom);

    edge_message<<<(int)(((size_t)N_EDGES * IN_DIM) / 256), 256, 0, stream>>>(
        edge_idx, h_ws, score_ws, denom, out);

    elu_kernel<<<(N_NODES * IN_DIM + 255) / 256, 256, 0, stream>>>(
        out, N_NODES * IN_DIM);
}